// Embedder_29300266893362
// MI455X (gfx1250) — compile-verified
//
#include <hip/hip_runtime.h>
#include <hip/hip_bf16.h>

// Embedder one-hot count: out[b, t] = sum_s (tokens[b,s] == t) * on_values[b,s]
// B=1024, S=200, DEPTH=100000. Pure memory-bound scatter-add:
//   - 409.6 MB output fill dominates (~17.6 us at 23.3 TB/s)
//   - 204,800 fp32 atomics in L2 are negligible
// Strategy: one block per batch row (row is block-private), vectorized
// non-temporal b128 zero-fill, then hardware global_atomic_add_f32 scatter.

typedef float float4v __attribute__((ext_vector_type(4)));

__global__ void embedder_count_kernel(const int* __restrict__ tokens,
                                      const float* __restrict__ on_values,
                                      float* __restrict__ out,
                                      int S, int depth) {
    const int b = blockIdx.x;
    float* __restrict__ row = out + (size_t)b * (size_t)depth;

    // ---- Phase 1: zero this row with 128-bit non-temporal stores ----
    // depth = 100000 -> 25000 exact float4 stores; row byte offset
    // (b * 400000) is 16B-aligned, so the vector path is always aligned.
    const int n4 = depth >> 2;
    float4v zero = {0.0f, 0.0f, 0.0f, 0.0f};
    float4v* __restrict__ row4 = (float4v*)row;
    for (int i = threadIdx.x; i < n4; i += blockDim.x) {
        __builtin_nontemporal_store(zero, row4 + i);   // global_store_b128 th:NT
    }
    // scalar tail (empty for depth % 4 == 0, kept for generality)
    for (int i = (n4 << 2) + (int)threadIdx.x; i < depth; i += blockDim.x) {
        __builtin_nontemporal_store(0.0f, row + i);
    }

    // Zero stores must be visible at L2 (where atomics execute) before any
    // wave of this block issues its scatter atomics.
    __threadfence();      // s_wait_storecnt 0 (+ device-scope release)
    __syncthreads();

    // ---- Phase 2: scatter-add this row's S tokens ----
    const int*   __restrict__ trow = tokens    + (size_t)b * (size_t)S;
    const float* __restrict__ vrow = on_values + (size_t)b * (size_t)S;
    for (int s = threadIdx.x; s < S; s += blockDim.x) {
        int   t = trow[s];
        float v = vrow[s];
        float* p = row + t;
        // CDNA5 hardware FP32 atomic add, no-return form (STOREcnt-tracked).
        asm volatile("global_atomic_add_f32 %0, %1, off"
                     :
                     : "v"(p), "v"(v)
                     : "memory");
    }
}

extern "C" void kernel_launch(void* const* d_in, const int* in_sizes, int n_in,
                              void* d_out, int out_size, void* d_ws, size_t ws_size,
                              hipStream_t stream) {
    (void)n_in; (void)d_ws; (void)ws_size;
    const int*   tokens    = (const int*)d_in[0];
    const float* on_values = (const float*)d_in[1];
    float*       out       = (float*)d_out;

    const int depth = 100000;                 // DEPTH from the reference
    int B = out_size / depth;                 // 1024
    if (B <= 0) B = 1;
    int S = in_sizes[0] / B;                  // 200

    dim3 grid(B);        // one block per output row -> row-private fill
    dim3 block(256);     // 8 wave32 waves per block

    embedder_count_kernel<<<grid, block, 0, stream>>>(tokens, on_values, out, S, depth);
}